// BEVConvSV10_19696720019820
// MI455X (gfx1250) — compile-verified
//
#include <hip/hip_runtime.h>
#include <hip/hip_bf16.h>
#include <math.h>

// ---------------- problem constants (from reference) ----------------
constexpr int W  = 1408;          // SIZE[0]
constexpr int H  = 1600;          // SIZE[1]
constexpr int B  = 4;             // batch_size
constexpr int HW = W * H;         // 2,252,800 per-channel plane
constexpr long long BEV_ELEMS = (long long)B * 4 * HW;   // 36,044,800
constexpr int Ho = H / 8;         // 200
constexpr int Wo = W / 8;         // 176
constexpr int NPOOL = B * Ho * Wo;                       // 140,800 (8x8 blocks, all ch)

// Scales exactly as the reference computes them (double math, then f32 use)
constexpr float X_SCALE = (float)(1408.0 / (70.4 - 0.0));   // == 20.0f
constexpr float Y_SCALE = (float)(1600.0 / (40.0 - -40.0)); // == 20.0f
constexpr float Y_OFF   = (float)((40.0 - -40.0) / 2.0);    // 40.0f

// ---------------- kernel 1: init the BEV accumulators in d_out ----------------
// layout: out[((b*4 + c)*H + y)*W + x]; ch0=cnt(bits as u32), ch1=imax,
// ch2=zsum, ch3=zmin(init 10.0). HW % 4 == 0 so each float4 stays in one channel.
__global__ void bev_init_kernel(float4* __restrict__ out, int nvec) {
    int i = blockIdx.x * blockDim.x + threadIdx.x;
    if (i >= nvec) return;
    constexpr int VEC_PER_CH = HW / 4;
    int c = (i / VEC_PER_CH) & 3;
    float v = (c == 3) ? 10.0f : 0.0f;
    out[i] = make_float4(v, v, v, v);
}

// ---------------- kernel 2: scatter points with native atomics ----------------
__global__ void bev_scatter_kernel(const float* __restrict__ pts,
                                   float* __restrict__ out, int n) {
    int i = blockIdx.x * blockDim.x + threadIdx.x;
    if (i >= n) return;
    const float* p = pts + (size_t)i * 5;
    float bf    = __builtin_nontemporal_load(p + 0);
    float x     = __builtin_nontemporal_load(p + 1);
    float y     = __builtin_nontemporal_load(p + 2);
    float z     = __builtin_nontemporal_load(p + 3);
    float inten = __builtin_nontemporal_load(p + 4);

    int xp = (int)(x * X_SCALE);                 // trunc toward zero == astype(int32)
    int yp = (int)((y + Y_OFF) * Y_SCALE);
    if (xp < 0 || xp >= W || yp < 0 || yp >= H) return;  // masked points are no-ops

    int bi = (int)bf;
    size_t cell = (size_t)yp * W + xp;
    float* base = out + (size_t)bi * 4 * HW + cell;

    // ch0: count (integer add on the f32 slot's bits; init 0.0f == 0)
    atomicAdd((unsigned int*)base, 1u);
    // ch1: max intensity; inten >= 0 so int-bit compare is monotonic; init 0.0f
    atomicMax((int*)(base + (size_t)HW), __float_as_int(inten));
    // ch2: z sum (native f32 atomic add)
    atomicAdd(base + 2 * (size_t)HW, z);
    // ch3: z min via sign-split bit trick (z may be negative); init 10.0f
    float* mp = base + 3 * (size_t)HW;
    if (z >= 0.0f) atomicMin((int*)mp, __float_as_int(z));
    else           atomicMax((unsigned int*)mp, __float_as_uint(z));
}

// ---------------- kernel 3: fused finalize (pts, zmean) + 8x8 max pool ----------------
__device__ __forceinline__ float csafe(int c) { return c ? (float)c : 1.0f; }
__device__ __forceinline__ float max4(float4 v) {
    return fmaxf(fmaxf(v.x, v.y), fmaxf(v.z, v.w));
}

__global__ void bev_finalize_pool_kernel(float* __restrict__ out,
                                         float* __restrict__ spatial, int n) {
    int i = blockIdx.x * blockDim.x + threadIdx.x;   // one thread per (b, ho, wo)
    if (i >= n) return;
    int wo = i % Wo;
    int t  = i / Wo;
    int ho = t % Ho;
    int b  = t / Ho;

    size_t base = (size_t)b * 4 * HW + (size_t)(ho * 8) * W + (size_t)(wo * 8);
    const size_t CH = (size_t)HW;

    float m0 = -INFINITY, m1 = -INFINITY, m2 = -INFINITY, m3 = -INFINITY;

    for (int r = 0; r < 8; ++r) {
        size_t rb = base + (size_t)r * W;
        // 8 cells of this row: two float4 groups per channel (16B aligned)
        int4   ca = *(const int4*)  (out + rb);              // ch0 cnt bits
        int4   cb = *(const int4*)  (out + rb + 4);
        float4 ia = *(const float4*)(out + rb + CH);         // ch1 imax
        float4 ib = *(const float4*)(out + rb + CH + 4);
        float4 za = *(const float4*)(out + rb + 2 * CH);     // ch2 zsum
        float4 zb = *(const float4*)(out + rb + 2 * CH + 4);
        float4 na = *(const float4*)(out + rb + 3 * CH);     // ch3 zmin
        float4 nb = *(const float4*)(out + rb + 3 * CH + 4);

        float c0 = csafe(ca.x), c1 = csafe(ca.y), c2 = csafe(ca.z), c3 = csafe(ca.w);
        float c4 = csafe(cb.x), c5 = csafe(cb.y), c6 = csafe(cb.z), c7 = csafe(cb.w);

        float4 pa = make_float4(c0 / 50.0f, c1 / 50.0f, c2 / 50.0f, c3 / 50.0f);
        float4 pb = make_float4(c4 / 50.0f, c5 / 50.0f, c6 / 50.0f, c7 / 50.0f);
        float4 ma = make_float4(za.x / c0, za.y / c1, za.z / c2, za.w / c3);
        float4 mb = make_float4(zb.x / c4, zb.y / c5, zb.z / c6, zb.w / c7);

        *(float4*)(out + rb)              = pa;   // finalized pts
        *(float4*)(out + rb + 4)          = pb;
        *(float4*)(out + rb + 2 * CH)     = ma;   // finalized zmean
        *(float4*)(out + rb + 2 * CH + 4) = mb;

        m0 = fmaxf(m0, fmaxf(max4(pa), max4(pb)));
        m1 = fmaxf(m1, fmaxf(max4(ia), max4(ib)));
        m2 = fmaxf(m2, fmaxf(max4(ma), max4(mb)));
        m3 = fmaxf(m3, fmaxf(max4(na), max4(nb)));
    }

    size_t so = ((size_t)(b * 4) * Ho + ho) * Wo + wo;
    const size_t PC = (size_t)Ho * Wo;
    spatial[so]          = m0;
    spatial[so + PC]     = m1;
    spatial[so + 2 * PC] = m2;
    spatial[so + 3 * PC] = m3;
}

// ---------------- launch ----------------
extern "C" void kernel_launch(void* const* d_in, const int* in_sizes, int n_in,
                              void* d_out, int out_size, void* d_ws, size_t ws_size,
                              hipStream_t stream) {
    const float* points = (const float*)d_in[0];
    int n = in_sizes[0] / 5;                  // 800,000 points
    float* out = (float*)d_out;               // [bev | spatial]
    float* spatial = out + BEV_ELEMS;

    const int TPB = 256;

    int init_vecs = (int)(BEV_ELEMS / 4);     // 9,011,200
    bev_init_kernel<<<(init_vecs + TPB - 1) / TPB, TPB, 0, stream>>>(
        (float4*)out, init_vecs);

    bev_scatter_kernel<<<(n + TPB - 1) / TPB, TPB, 0, stream>>>(points, out, n);

    bev_finalize_pool_kernel<<<(NPOOL + TPB - 1) / TPB, TPB, 0, stream>>>(
        out, spatial, NPOOL);
}